// GnnMapping_16338055594477
// MI455X (gfx1250) — compile-verified
//
#include <hip/hip_runtime.h>

#define N_QUBIT     20
#define N_LAYERP1   51
#define FEAT        128
#define BATCH       128
#define NODES_PER_B (N_QUBIT * N_LAYERP1)            // 1020
#define N_NODES     (BATCH * NODES_PER_B)            // 130560
#define N_EDGES     524288
#define NPAIR       (BATCH * N_QUBIT * N_QUBIT)      // 51200
#define NEG_SLOPE   0.01f

typedef __attribute__((ext_vector_type(2))) float v2f;
typedef __attribute__((ext_vector_type(8))) float v8f;

__device__ __forceinline__ float leaky(float v) {
  return v >= 0.0f ? v : NEG_SLOPE * v;
}

// ---------------- embedding gather: h[n] = emb[x[n]] ----------------
__global__ void k_embed(const int* __restrict__ x, const float* __restrict__ emb,
                        float* __restrict__ h) {
  int t = blockIdx.x * blockDim.x + threadIdx.x;   // (node, quad-of-4-floats)
  int node = t >> 5;
  int f4 = (t & 31) << 2;
  if (node >= N_NODES) return;
  float4 v = *(const float4*)(emb + (size_t)x[node] * FEAT + f4);
  *(float4*)(h + (size_t)node * FEAT + f4) = v;
}

// ---------------- degree / D^-1/2 (graph is static; computed once) ----------------
__global__ void k_deg_init(float* __restrict__ deg) {
  int i = blockIdx.x * blockDim.x + threadIdx.x;
  if (i < N_NODES) deg[i] = 1.0f;                  // self-loop
}
__global__ void k_deg_edges(const int* __restrict__ dst, float* __restrict__ deg) {
  int e = blockIdx.x * blockDim.x + threadIdx.x;
  if (e < N_EDGES) unsafeAtomicAdd(&deg[dst[e]], 1.0f);
}
__global__ void k_dinv(float* __restrict__ deg) {
  int i = blockIdx.x * blockDim.x + threadIdx.x;
  if (i < N_NODES) deg[i] = rsqrtf(deg[i]);        // deg >= 1 always (self-loop)
}

// ---------------- generic f32 WMMA GEMM: out = act(A[MxK] @ W[KxN] + bias) --------
// block = 256 threads = 8 wave32s; each wave -> 32 rows x 64 cols; block -> 256 rows.
// W slab staged into LDS with GLOBAL_LOAD_ASYNC_TO_LDS_B128 (ASYNCcnt DMA path);
// A fragments software-pipelined to keep v_wmma off the loadcnt critical path.
__launch_bounds__(256)
__global__ void k_wmma_gemm(const float* __restrict__ A, const float* __restrict__ W,
                            const float* __restrict__ bias, float* __restrict__ out,
                            int M, int K, int N, int act) {
  __shared__ float wtile[128 * 64];                // K-chunk x 64-col slab of W (32 KB)
  const int lane = threadIdx.x & 31;
  const int wave = threadIdx.x >> 5;
  const int half = lane >> 4;                      // lane group (0: lanes 0-15, 1: 16-31)
  const int ln   = lane & 15;
  const int rowbase = blockIdx.x * 256 + wave * 32;
  const int n0 = blockIdx.y * 64;
  const unsigned lds_base = (unsigned)(uintptr_t)&wtile[0];  // generic->LDS offset (addr[31:0])

  v8f acc[2][4] = {};                              // 2 m-subtiles x 4 n-subtiles (16x16 f32)

  for (int kc = 0; kc < K; kc += 128) {
    const int kchunk = (K - kc < 128) ? (K - kc) : 128;
    // ---- async DMA stage: each lane moves 16B global -> LDS, no VGPR round-trip
    for (int q = threadIdx.x; q < kchunk * 16; q += 256) {
      const int fi = q << 2;                       // float index into slab
      const int k = fi >> 6, n = fi & 63;
      const float* gsrc = W + (size_t)(kc + k) * N + n0 + n;
      const unsigned ldst = lds_base + ((unsigned)fi << 2);
      asm volatile("global_load_async_to_lds_b128 %0, %1, off"
                   :: "v"(ldst), "v"(gsrc) : "memory");
    }
    asm volatile("s_wait_asynccnt 0x0" ::: "memory");
    __syncthreads();

    if (rowbase < M) {
      const float* arow0 = A + (size_t)(rowbase + ln) * K + kc;
      const float* arow1 = arow0 + (size_t)16 * K;
      const int kaoff = half * 2;
      // prime the A-fragment pipeline
      float2 a0 = *(const float2*)(arow0 + kaoff);
      float2 a1 = *(const float2*)(arow1 + kaoff);
      for (int k0 = 0; k0 < kchunk; k0 += 4) {
        float2 na0 = a0, na1 = a1;
        if (k0 + 4 < kchunk) {                     // prefetch next k-step's fragments
          na0 = *(const float2*)(arow0 + k0 + 4 + kaoff);
          na1 = *(const float2*)(arow1 + k0 + 4 + kaoff);
        }
        // A-fragment (16x4 f32): VGPR0 = {K0 | K2}, VGPR1 = {K1 | K3} by lane half
        v2f av0; av0.x = a0.x; av0.y = a0.y;
        v2f av1; av1.x = a1.x; av1.y = a1.y;
        const int ka = k0 + kaoff;
#pragma unroll
        for (int t = 0; t < 4; ++t) {
          // B-fragment (4x16 f32): row striped across lanes, mirrored K split;
          // reused by both m-subtiles (2 WMMAs per DS fetch)
          v2f b;
          b.x = wtile[ka * 64 + t * 16 + ln];
          b.y = wtile[(ka + 1) * 64 + t * 16 + ln];
          acc[0][t] = __builtin_amdgcn_wmma_f32_16x16x4_f32(
              false, av0, false, b, (short)0, acc[0][t], false, false);
          acc[1][t] = __builtin_amdgcn_wmma_f32_16x16x4_f32(
              false, av1, false, b, (short)0, acc[1][t], false, false);
        }
        a0 = na0; a1 = na1;
      }
    }
    __syncthreads();
  }
  if (rowbase >= M) return;
#pragma unroll
  for (int mt = 0; mt < 2; ++mt) {
    const int rb = rowbase + mt * 16;
#pragma unroll
    for (int t = 0; t < 4; ++t) {
      const int col = n0 + t * 16 + ln;
      const float bv = bias ? bias[col] : 0.0f;
#pragma unroll
      for (int r = 0; r < 8; ++r) {
        // C/D layout: VGPR r holds row r (lanes 0-15) / row r+8 (lanes 16-31)
        const int row = rb + half * 8 + r;
        float v = acc[mt][t][r] + bv;
        if (act) v = leaky(v);
        out[(size_t)row * N + col] = v;
      }
    }
  }
}

// ---------------- GCN aggregation ----------------
// self-loop seed: agg[i] = hw[i] * dinv[i]^2
__global__ void k_selfloop(const float* __restrict__ hw, const float* __restrict__ dinv,
                           float* __restrict__ agg) {
  int t = blockIdx.x * blockDim.x + threadIdx.x;
  int node = t >> 5;
  int f4 = (t & 31) << 2;
  if (node >= N_NODES) return;
  float d = dinv[node];
  float s = d * d;
  float4 v = *(const float4*)(hw + (size_t)node * FEAT + f4);
  v.x *= s; v.y *= s; v.z *= s; v.w *= s;
  *(float4*)(agg + (size_t)node * FEAT + f4) = v;
}

// edge scatter: agg[dst] += hw[src] * dinv[src]*dinv[dst]  (L2-resident f32 atomics)
__global__ void k_edge(const float* __restrict__ hw, const int* __restrict__ src,
                       const int* __restrict__ dst, const float* __restrict__ dinv,
                       float* __restrict__ agg) {
  int t = blockIdx.x * blockDim.x + threadIdx.x;   // (edge, quad)
  int e = t >> 5;
  int f4 = (t & 31) << 2;
  if (e >= N_EDGES) return;
  int s = src[e], d = dst[e];
  float norm = dinv[s] * dinv[d];
  float4 v = *(const float4*)(hw + (size_t)s * FEAT + f4);
  float* o = agg + (size_t)d * FEAT + f4;
  unsafeAtomicAdd(o + 0, v.x * norm);
  unsafeAtomicAdd(o + 1, v.y * norm);
  unsafeAtomicAdd(o + 2, v.z * norm);
  unsafeAtomicAdd(o + 3, v.w * norm);
}

// combine: h = (act ? leaky(agg + b) : (agg + b)) + h
__global__ void k_combine(const float* __restrict__ agg, const float* __restrict__ bias,
                          float* __restrict__ h, int act) {
  int t = blockIdx.x * blockDim.x + threadIdx.x;
  int node = t >> 5;
  int f4 = (t & 31) << 2;
  if (node >= N_NODES) return;
  float4 a = *(const float4*)(agg + (size_t)node * FEAT + f4);
  float4 hv = *(float4*)(h + (size_t)node * FEAT + f4);
  a.x += bias[f4 + 0]; a.y += bias[f4 + 1]; a.z += bias[f4 + 2]; a.w += bias[f4 + 3];
  if (act) { a.x = leaky(a.x); a.y = leaky(a.y); a.z = leaky(a.z); a.w = leaky(a.w); }
  hv.x += a.x; hv.y += a.y; hv.z += a.z; hv.w += a.w;
  *(float4*)(h + (size_t)node * FEAT + f4) = hv;
}

// ---------------- pair assembly: p0[pair] = concat(h[b,i], h[b,j]) ----------------
__global__ void k_pairs(const float* __restrict__ h, float* __restrict__ p0) {
  int t = blockIdx.x * blockDim.x + threadIdx.x;   // (pair, quad-of-256)
  int pair = t >> 6;
  int f4 = (t & 63) << 2;
  if (pair >= NPAIR) return;
  int b = pair / (N_QUBIT * N_QUBIT);
  int r = pair - b * (N_QUBIT * N_QUBIT);
  int i = r / N_QUBIT, j = r - (r / N_QUBIT) * N_QUBIT;
  int node = b * NODES_PER_B + ((f4 < FEAT) ? i : j);
  int fo = f4 & (FEAT - 1);
  *(float4*)(p0 + (size_t)pair * 256 + f4) =
      *(const float4*)(h + (size_t)node * FEAT + fo);
}

// ---------------- final 64->1 dot + symmetrize ----------------
__global__ void k_final(const float* __restrict__ p, const float* __restrict__ w,
                        const float* __restrict__ b, float* __restrict__ scores) {
  int pair = blockIdx.x * blockDim.x + threadIdx.x;
  if (pair >= NPAIR) return;
  const float* row = p + (size_t)pair * 64;
  float s = 0.0f;
#pragma unroll
  for (int k = 0; k < 64; ++k) s += row[k] * w[k];
  scores[pair] = s + b[0];
}

__global__ void k_sym(const float* __restrict__ scores, float* __restrict__ out) {
  int t = blockIdx.x * blockDim.x + threadIdx.x;
  if (t >= NPAIR) return;
  int b = t / (N_QUBIT * N_QUBIT);
  int r = t - b * (N_QUBIT * N_QUBIT);
  int i = r / N_QUBIT, j = r - (r / N_QUBIT) * N_QUBIT;
  float a0 = scores[b * 400 + i * N_QUBIT + j];
  float a1 = scores[b * 400 + j * N_QUBIT + i];
  out[t] = 0.5f * (a0 + a1);
}

extern "C" void kernel_launch(void* const* d_in, const int* in_sizes, int n_in,
                              void* d_out, int out_size, void* d_ws, size_t ws_size,
                              hipStream_t stream) {
  (void)in_sizes; (void)n_in; (void)out_size; (void)ws_size;
  const int*   x     = (const int*)d_in[0];
  const int*   eidx  = (const int*)d_in[1];
  const int*   src   = eidx;
  const int*   dst   = eidx + N_EDGES;
  const float* emb   = (const float*)d_in[2];
  const float* gcn_w = (const float*)d_in[3];
  const float* gcn_b = (const float*)d_in[4];
  const float* mw[6]; const float* mb[6];
  for (int i = 0; i < 6; ++i) {
    mw[i] = (const float*)d_in[5 + 2 * i];
    mb[i] = (const float*)d_in[6 + 2 * i];
  }

  char* ws = (char*)d_ws;
  const size_t HB = (size_t)N_NODES * FEAT * sizeof(float);     // 66.85 MB
  float* h      = (float*)(ws);
  float* hw     = (float*)(ws + HB);
  float* agg    = (float*)(ws + 2 * HB);
  float* dinv   = (float*)(ws + 3 * HB);
  float* scores = (float*)(ws + 3 * HB + ((size_t)N_NODES * 4 + 1024));
  float* p0 = hw;   // GCN scratch reused for pair MLP ping-pong
  float* p1 = agg;

  const int TB = 256;
  // 1) embedding gather
  k_embed<<<(N_NODES * 32 + TB - 1) / TB, TB, 0, stream>>>(x, emb, h);
  // 2) degree -> D^-1/2 (once)
  k_deg_init<<<(N_NODES + TB - 1) / TB, TB, 0, stream>>>(dinv);
  k_deg_edges<<<(N_EDGES + TB - 1) / TB, TB, 0, stream>>>(dst, dinv);
  k_dinv<<<(N_NODES + TB - 1) / TB, TB, 0, stream>>>(dinv);
  // 3) five GCN layers
  for (int L = 0; L < 5; ++L) {
    dim3 g(N_NODES / 256, FEAT / 64);
    k_wmma_gemm<<<g, TB, 0, stream>>>(h, gcn_w + (size_t)L * FEAT * FEAT, nullptr,
                                      hw, N_NODES, FEAT, FEAT, 0);
    k_selfloop<<<(N_NODES * 32 + TB - 1) / TB, TB, 0, stream>>>(hw, dinv, agg);
    k_edge<<<(N_EDGES * 32 + TB - 1) / TB, TB, 0, stream>>>(hw, src, dst, dinv, agg);
    k_combine<<<(N_NODES * 32 + TB - 1) / TB, TB, 0, stream>>>(
        agg, gcn_b + (size_t)L * FEAT, h, (L < 4) ? 1 : 0);
  }
  // 4) pair features [51200 x 256]
  k_pairs<<<(NPAIR * 64 + TB - 1) / TB, TB, 0, stream>>>(h, p0);
  // 5) pair MLP (WMMA GEMMs, fused bias+leaky)
  { dim3 g(NPAIR / 256, 256 / 64);
    k_wmma_gemm<<<g, TB, 0, stream>>>(p0, mw[0], mb[0], p1, NPAIR, 256, 256, 1); }
  { dim3 g(NPAIR / 256, 128 / 64);
    k_wmma_gemm<<<g, TB, 0, stream>>>(p1, mw[1], mb[1], p0, NPAIR, 256, 128, 1); }
  { dim3 g(NPAIR / 256, 128 / 64);
    k_wmma_gemm<<<g, TB, 0, stream>>>(p0, mw[2], mb[2], p1, NPAIR, 128, 128, 1); }
  { dim3 g(NPAIR / 256, 64 / 64);
    k_wmma_gemm<<<g, TB, 0, stream>>>(p1, mw[3], mb[3], p0, NPAIR, 128, 64, 1); }
  { dim3 g(NPAIR / 256, 64 / 64);
    k_wmma_gemm<<<g, TB, 0, stream>>>(p0, mw[4], mb[4], p1, NPAIR, 64, 64, 1); }
  // 6) 64->1 head + symmetrize
  k_final<<<(NPAIR + TB - 1) / TB, TB, 0, stream>>>(p1, mw[5], mb[5], scores);
  k_sym<<<(NPAIR + TB - 1) / TB, TB, 0, stream>>>(scores, (float*)d_out);
}